// Encoder_44736379355603
// MI455X (gfx1250) — compile-verified
//
#include <hip/hip_runtime.h>
#include <hip/hip_bf16.h>

typedef __attribute__((ext_vector_type(2))) float v2f;
typedef __attribute__((ext_vector_type(8))) float v8f;

#define IN_C  128
#define HID_C 64
#define OUT_C 16
#define CAT_C 32   // mu|logstd fused

// ---------------- degree / normalization ----------------

__global__ void k_init_deg(float* deg, int n) {
    int i = blockIdx.x * blockDim.x + threadIdx.x;
    if (i < n) deg[i] = 1.0f;   // self loop
}

__global__ void k_count_deg(const long long* __restrict__ dst, float* deg, long long E) {
    long long e = (long long)blockIdx.x * blockDim.x + threadIdx.x;
    if (e >= E) return;
    __builtin_prefetch(dst + e + 2048, 0, 1);
    atomicAdd(&deg[(int)dst[e]], 1.0f);
}

__global__ void k_deg_to_dis(float* deg, int n) {
    int i = blockIdx.x * blockDim.x + threadIdx.x;
    if (i < n) deg[i] = rsqrtf(deg[i]);   // deg >= 1 always (self loop)
}

// ---------------- GEMM1: hp = dis .* (x @ W1), seeds acc with self-loop ----------------
// One 16x16 output tile per wave; 4 waves/block cover the 64 output columns.
// A (16x4 f32): lane L holds A[M=L%16, K=2*(L/16)+j] in element j.
// B (4x16 f32): lane L holds B[K=2*(L/16)+j, N=L%16] in element j.
// C/D: VGPR r -> lanes 0-15 row r, lanes 16-31 row r+8.

__global__ __launch_bounds__(128) void k_gemm1_wmma(
    const float* __restrict__ x,    // [N,128]
    const float* __restrict__ W1,   // [128,64]
    const float* __restrict__ dis,  // [N]
    float* __restrict__ hp,         // [N,64]  dis-scaled x@W1
    float* __restrict__ acc,        // [N,64]  accumulator (init = hp)
    int n)
{
    const int wave = threadIdx.x >> 5;       // col tile 0..3
    const int lane = threadIdx.x & 31;
    const int half = lane >> 4;              // 0/1
    const int m    = lane & 15;
    const int row0 = blockIdx.x * 16;
    const int col0 = wave * 16;

    const int arow = min(row0 + m, n - 1);
    const float* ap = x  + (size_t)arow * IN_C + half * 2;
    const float* bp = W1 + (size_t)(half * 2) * HID_C + col0 + m;

    v8f c = {0.f,0.f,0.f,0.f,0.f,0.f,0.f,0.f};
    #pragma unroll
    for (int k = 0; k < IN_C; k += 4) {
        v2f a, b;
        a.x = ap[k];
        a.y = ap[k + 1];
        b.x = bp[(size_t)k * HID_C];
        b.y = bp[(size_t)(k + 1) * HID_C];
        c = __builtin_amdgcn_wmma_f32_16x16x4_f32(
                false, a, false, b, (short)0, c, false, false);
    }

    #pragma unroll
    for (int r = 0; r < 8; ++r) {
        int row = row0 + r + half * 8;
        if (row < n) {
            float v = c[r] * dis[row];          // pre-scale row by D^-1/2
            size_t o = (size_t)row * HID_C + col0 + m;
            hp[o]  = v;
            acc[o] = v;                          // self-loop contribution
        }
    }
}

// ---------------- scatter: acc[dst] += hp[src], 64 channels ----------------
__global__ void k_scatter64(const long long* __restrict__ src,
                            const long long* __restrict__ dst,
                            const float* __restrict__ hp,
                            float* __restrict__ acc, long long E)
{
    long long t = (long long)blockIdx.x * blockDim.x + threadIdx.x;
    long long e = t >> 4;              // 16 threads per edge, 4 ch each
    int cg = (int)(t & 15);
    if (e >= E) return;
    __builtin_prefetch(src + e + 512, 0, 1);
    __builtin_prefetch(dst + e + 512, 0, 1);
    int s = (int)src[e], d = (int)dst[e];
    const float4 v = *(const float4*)(hp + (size_t)s * HID_C + cg * 4);
    float* p = acc + (size_t)d * HID_C + cg * 4;
    atomicAdd(p + 0, v.x);
    atomicAdd(p + 1, v.y);
    atomicAdd(p + 2, v.z);
    atomicAdd(p + 3, v.w);
}

// ---------------- finalize1: ht = dis .* relu(dis .* acc + b1) ----------------
__global__ void k_finalize1(const float* __restrict__ acc,
                            const float* __restrict__ dis,
                            const float* __restrict__ b1,
                            float* __restrict__ ht, int n)
{
    long long idx = (long long)blockIdx.x * blockDim.x + threadIdx.x;
    if (idx >= (long long)n * HID_C) return;
    int i = (int)(idx >> 6);
    int ch = (int)(idx & 63);
    float di = dis[i];
    float v = fmaxf(di * acc[idx] + b1[ch], 0.0f);
    ht[idx] = di * v;                  // pre-scale for the next aggregation
}

// ---------------- GEMM2: P = ht @ [W_mu | W_ls] ([64,32]), seeds acc2 ----------------
__global__ __launch_bounds__(64) void k_gemm2_wmma(
    const float* __restrict__ ht,    // [N,64]
    const float* __restrict__ Wmu,   // [64,16]
    const float* __restrict__ Wls,   // [64,16]
    float* __restrict__ P,           // [N,32]
    float* __restrict__ acc2,        // [N,32]
    int n)
{
    const int wave = threadIdx.x >> 5;       // 0 -> mu cols, 1 -> logstd cols
    const int lane = threadIdx.x & 31;
    const int half = lane >> 4;
    const int m    = lane & 15;
    const int row0 = blockIdx.x * 16;
    const int col0 = wave * 16;

    const float* Wb = (wave == 0) ? Wmu : Wls;   // each [64,16] row-major
    const int arow = min(row0 + m, n - 1);
    const float* ap = ht + (size_t)arow * HID_C + half * 2;
    const float* bp = Wb + (size_t)(half * 2) * OUT_C + m;

    v8f c = {0.f,0.f,0.f,0.f,0.f,0.f,0.f,0.f};
    #pragma unroll
    for (int k = 0; k < HID_C; k += 4) {
        v2f a, b;
        a.x = ap[k];
        a.y = ap[k + 1];
        b.x = bp[(size_t)k * OUT_C];
        b.y = bp[(size_t)(k + 1) * OUT_C];
        c = __builtin_amdgcn_wmma_f32_16x16x4_f32(
                false, a, false, b, (short)0, c, false, false);
    }

    #pragma unroll
    for (int r = 0; r < 8; ++r) {
        int row = row0 + r + half * 8;
        if (row < n) {
            float v = c[r];
            size_t o = (size_t)row * CAT_C + col0 + m;
            P[o]    = v;
            acc2[o] = v;                         // self-loop contribution
        }
    }
}

// ---------------- scatter: acc2[dst] += P[src], 32 channels ----------------
__global__ void k_scatter32(const long long* __restrict__ src,
                            const long long* __restrict__ dst,
                            const float* __restrict__ P,
                            float* __restrict__ acc2, long long E)
{
    long long t = (long long)blockIdx.x * blockDim.x + threadIdx.x;
    long long e = t >> 3;              // 8 threads per edge, 4 ch each
    int cg = (int)(t & 7);
    if (e >= E) return;
    __builtin_prefetch(src + e + 1024, 0, 1);
    __builtin_prefetch(dst + e + 1024, 0, 1);
    int s = (int)src[e], d = (int)dst[e];
    const float4 v = *(const float4*)(P + (size_t)s * CAT_C + cg * 4);
    float* p = acc2 + (size_t)d * CAT_C + cg * 4;
    atomicAdd(p + 0, v.x);
    atomicAdd(p + 1, v.y);
    atomicAdd(p + 2, v.z);
    atomicAdd(p + 3, v.w);
}

// ---------------- finalize2: split into mu / logstd with biases ----------------
__global__ void k_finalize2(const float* __restrict__ acc2,
                            const float* __restrict__ dis,
                            const float* __restrict__ bmu,
                            const float* __restrict__ bls,
                            float* __restrict__ out, int n)
{
    long long idx = (long long)blockIdx.x * blockDim.x + threadIdx.x;
    if (idx >= (long long)n * CAT_C) return;
    int i = (int)(idx >> 5);
    int ch = (int)(idx & 31);
    float v = dis[i] * acc2[idx];
    if (ch < OUT_C)
        out[(size_t)i * OUT_C + ch] = v + bmu[ch];                       // mu
    else
        out[(size_t)n * OUT_C + (size_t)i * OUT_C + (ch - OUT_C)]
            = v + bls[ch - OUT_C];                                        // logstd
}

// ---------------- launch ----------------

extern "C" void kernel_launch(void* const* d_in, const int* in_sizes, int n_in,
                              void* d_out, int out_size, void* d_ws, size_t ws_size,
                              hipStream_t stream) {
    const float*     x   = (const float*)d_in[0];
    const long long* ei  = (const long long*)d_in[1];   // [2, E] int64
    const float*     W1  = (const float*)d_in[2];
    const float*     b1  = (const float*)d_in[3];
    const float*     Wmu = (const float*)d_in[4];
    const float*     bmu = (const float*)d_in[5];
    const float*     Wls = (const float*)d_in[6];
    const float*     bls = (const float*)d_in[7];
    float* out = (float*)d_out;

    const int       n = in_sizes[0] / IN_C;       // 100000
    const long long E = (long long)in_sizes[1] / 2;
    const long long* src = ei;
    const long long* dst = ei + E;

    // workspace layout (floats): dis[n] | bufA[n*64] | bufB[n*64]
    float* dis  = (float*)d_ws;
    float* bufA = dis + n;                  // hp, later ht
    float* bufB = bufA + (size_t)n * HID_C; // acc1, later P[n*32] | acc2[n*32]
    float* P    = bufB;
    float* acc2 = bufB + (size_t)n * CAT_C;

    const int rowTiles = (n + 15) / 16;

    // 1) degree -> deg_inv_sqrt
    k_init_deg<<<(n + 255) / 256, 256, 0, stream>>>(dis, n);
    k_count_deg<<<(unsigned)((E + 255) / 256), 256, 0, stream>>>(dst, dis, E);
    k_deg_to_dis<<<(n + 255) / 256, 256, 0, stream>>>(dis, n);

    // 2) hp = dis .* (x @ W1); acc1 seeded with self loop
    k_gemm1_wmma<<<rowTiles, 128, 0, stream>>>(x, W1, dis, bufA, bufB, n);

    // 3) 64-channel edge scatter into acc1
    {
        long long threads = E * 16;
        k_scatter64<<<(unsigned)((threads + 255) / 256), 256, 0, stream>>>(
            src, dst, bufA, bufB, E);
    }

    // 4) ht = dis .* relu(dis .* acc1 + b1)   (written over bufA)
    {
        long long threads = (long long)n * HID_C;
        k_finalize1<<<(unsigned)((threads + 255) / 256), 256, 0, stream>>>(
            bufB, dis, b1, bufA, n);
    }

    // 5) P = ht @ [Wmu|Wls]; acc2 seeded with self loop
    k_gemm2_wmma<<<rowTiles, 64, 0, stream>>>(bufA, Wmu, Wls, P, acc2, n);

    // 6) 32-channel edge scatter into acc2
    {
        long long threads = E * 8;
        k_scatter32<<<(unsigned)((threads + 255) / 256), 256, 0, stream>>>(
            src, dst, P, acc2, E);
    }

    // 7) mu / logstd
    {
        long long threads = (long long)n * CAT_C;
        k_finalize2<<<(unsigned)((threads + 255) / 256), 256, 0, stream>>>(
            acc2, dis, bmu, bls, out, n);
    }
}